// EnergyModel_79379585565534
// MI455X (gfx1250) — compile-verified
//
#include <hip/hip_runtime.h>
#include <math.h>

#define B_     2
#define NLOC_  2048
#define NALL_  4096
#define NNEI_  64
#define NA_    (B_*NLOC_)
#define EPS_   1e-8f
#define RCUT_  6.0f
#define RSMTH_ 0.5f

typedef float v2f __attribute__((ext_vector_type(2)));
typedef float v8f __attribute__((ext_vector_type(8)));

// ---------------------------------------------------------------- switch fn
__device__ __forceinline__ float sw_fn(float r, float* dsw) {
    float inv = 1.0f / fmaxf(r, EPS_);
    if (r < RSMTH_) { *dsw = -inv * inv; return inv; }
    if (r < RCUT_) {
        const float denom = 1.0f / (RCUT_ - RSMTH_);
        float uu  = (r - RSMTH_) * denom;
        float uu2 = uu * uu;
        float poly  = uu * uu2 * (-6.0f * uu2 + 15.0f * uu - 10.0f) + 1.0f;
        float um1   = uu - 1.0f;
        float dpoly = -30.0f * uu2 * um1 * um1;
        *dsw = dpoly * denom * inv - poly * inv * inv;
        return poly * inv;
    }
    *dsw = 0.0f;
    return 0.0f;
}

// ---------------------------------------------------------------- utilities
__global__ void kzero(float* p, int n) {
    int i = blockIdx.x * blockDim.x + threadIdx.x;
    if (i < n) p[i] = 0.0f;
}

__global__ void kext(const float* __restrict__ coord, const float* __restrict__ shift,
                     const int* __restrict__ mapping, float* __restrict__ ext) {
    int idx = blockIdx.x * blockDim.x + threadIdx.x;
    if (idx >= B_ * NALL_) return;
    int b = idx / NALL_;
    int m = mapping[idx];
    #pragma unroll
    for (int c = 0; c < 3; ++c)
        ext[idx * 3 + c] = coord[(b * NLOC_ + m) * 3 + c] - shift[idx * 3 + c];
}

// ------------------------------------------------ env matrix + embedding fwd
// block = 64 threads (1 thread per neighbor), grid = NA_
__global__ void kenv(const float* __restrict__ ext, const int* __restrict__ selected,
                     const float* __restrict__ ew1, const float* __restrict__ eb1,
                     const float* __restrict__ ew2, const float* __restrict__ eb2,
                     const float* __restrict__ ew3, const float* __restrict__ eb3,
                     float* __restrict__ GRg, float* __restrict__ Dg) {
    __shared__ float Gs[NNEI_ * 101];   // pad to 101 -> conflict-free column reads
    __shared__ float Rm[NNEI_ * 4];
    __shared__ float GRs[400];
    __shared__ float sw3[50 * 100];     // LDS-resident embedding weights (broadcast reads)
    __shared__ float sw2[25 * 50];
    __shared__ float sw1[25], sb1[25], sb2[50], sb3[100];

    int atom = blockIdx.x;
    int b = atom / NLOC_, i = atom % NLOC_;
    int t = threadIdx.x;

    for (int q = t; q < 5000; q += NNEI_) sw3[q] = ew3[q];
    for (int q = t; q < 1250; q += NNEI_) sw2[q] = ew2[q];
    if (t < 25) { sw1[t] = ew1[t]; sb1[t] = eb1[t]; }
    if (t < 50) sb2[t] = eb2[t];
    for (int q = t; q < 100; q += NNEI_) sb3[q] = eb3[q];

    int j = selected[(size_t)atom * NNEI_ + t];
    float cx = ext[((size_t)b * NALL_ + i) * 3 + 0];
    float cy = ext[((size_t)b * NALL_ + i) * 3 + 1];
    float cz = ext[((size_t)b * NALL_ + i) * 3 + 2];
    float x = ext[((size_t)b * NALL_ + j) * 3 + 0] - cx;
    float y = ext[((size_t)b * NALL_ + j) * 3 + 1] - cy;
    float z = ext[((size_t)b * NALL_ + j) * 3 + 2] - cz;
    float r = sqrtf(x * x + y * y + z * z + EPS_);
    float dsw;
    float s = sw_fn(r, &dsw);
    float invr = 1.0f / r;
    Rm[t * 4 + 0] = s;
    Rm[t * 4 + 1] = s * x * invr;
    Rm[t * 4 + 2] = s * y * invr;
    Rm[t * 4 + 3] = s * z * invr;
    __syncthreads();

    float h1[25], h2[50];
    #pragma unroll
    for (int a = 0; a < 25; ++a) h1[a] = tanhf(s * sw1[a] + sb1[a]);
    for (int a = 0; a < 50; ++a) {
        float p = sb2[a];
        #pragma unroll
        for (int q = 0; q < 25; ++q) p += h1[q] * sw2[q * 50 + a];
        h2[a] = tanhf(p) + h1[a % 25];
    }
    for (int jg = 0; jg < 100; ++jg) {
        float p = sb3[jg];
        #pragma unroll
        for (int q = 0; q < 50; ++q) p += h2[q] * sw3[q * 100 + jg];
        Gs[t * 101 + jg] = tanhf(p) + h2[jg % 50];
    }
    __syncthreads();

    // GR[c][m] = (1/NNEI) * sum_nei Rm[nei][c] * G[nei][m]
    for (int idx = t; idx < 400; idx += NNEI_) {
        int c = idx / 100, mm = idx % 100;
        float a = 0.0f;
        #pragma unroll 8
        for (int n = 0; n < NNEI_; ++n) a += Rm[n * 4 + c] * Gs[n * 101 + mm];
        a *= (1.0f / NNEI_);
        GRs[idx] = a;
        GRg[(size_t)atom * 400 + idx] = a;
    }
    __syncthreads();

    // D[m][l] = sum_c GR[c][m] * GR[c][l], l < 16
    for (int idx = t; idx < 1600; idx += NNEI_) {
        int mm = idx / 16, l = idx % 16;
        float d = 0.0f;
        #pragma unroll
        for (int c = 0; c < 4; ++c) d += GRs[c * 100 + mm] * GRs[c * 100 + l];
        Dg[(size_t)atom * 1600 + idx] = d;
    }
}

// ------------------------------------------------------- fp32 WMMA GEMM
// C[M,N] = A[M,K] @ B + epilogue ; Bmat(k,n) = TRANSB ? W[n*K+k] : W[k*N+n]
// MODE 0: C = acc
// MODE 1: C = tanh(acc + bias[n])
// MODE 2: C = tanh(acc + bias[n]) + skip[m,n]
// MODE 3: C = (acc + addvec?[n]) * (1 - (t1[m,n] - t2?[m,n])^2)   (tanh backward)
template <bool TRANSB, int MODE>
__global__ void wmma_gemm(const float* __restrict__ A, const float* __restrict__ W,
                          float* __restrict__ C,
                          const float* __restrict__ bias, const float* __restrict__ skip,
                          const float* __restrict__ t1, const float* __restrict__ t2,
                          const float* __restrict__ addvec,
                          int Mrows, int N, int K) {
    int lane = threadIdx.x;                       // 0..31
    int tile = blockIdx.x * blockDim.y + threadIdx.y;
    int ntN  = N >> 4;
    int ntot = (Mrows >> 4) * ntN;
    if (tile >= ntot) return;
    int tm = tile / ntN, tn = tile % ntN;
    int half = lane >> 4;                         // K-pair select (0 or 1)
    int l16  = lane & 15;
    int arow = tm * 16 + l16;                     // A: M row per lane
    int bcol = tn * 16 + l16;                     // B: N col per lane

    const float* ap = A + (size_t)arow * K + half * 2;
    const float* bp;
    size_t bstep;
    if (TRANSB) { bp = W + (size_t)bcol * K + half * 2;  bstep = 4; }
    else        { bp = W + (size_t)(half * 2) * N + bcol; bstep = (size_t)4 * N; }

    v8f acc = {0.f, 0.f, 0.f, 0.f, 0.f, 0.f, 0.f, 0.f};
    #pragma unroll 4
    for (int kk = 0; kk < K; kk += 4) {
        v2f a = *(const v2f*)ap;                  // 8B-aligned (K even, half*2 even)
        v2f bv;
        if (TRANSB) {
            bv = *(const v2f*)bp;
        } else {
            bv.x = bp[0];
            bv.y = bp[N];
        }
        acc = __builtin_amdgcn_wmma_f32_16x16x4_f32(false, a, false, bv,
                                                    (short)0, acc, false, false);
        ap += 4;
        bp += bstep;
    }
    #pragma unroll
    for (int i = 0; i < 8; ++i) {
        int row = tm * 16 + i + half * 8;
        int col = tn * 16 + l16;
        size_t o = (size_t)row * N + col;
        float v = acc[i];
        if constexpr (MODE == 1) {
            v = tanhf(v + bias[col]);
        } else if constexpr (MODE == 2) {
            v = tanhf(v + bias[col]) + skip[o];
        } else if constexpr (MODE == 3) {
            if (addvec) v += addvec[col];
            float tt = t1[o] - (t2 ? t2[o] : 0.0f);
            v *= (1.0f - tt * tt);
        }
        C[o] = v;
    }
}

// --------------------------------------------------------------- energy
__global__ void kenergy(const float* __restrict__ f3, const float* __restrict__ fw4,
                        const float* __restrict__ fb4, const float* __restrict__ bias_atom,
                        const int* __restrict__ atype, float* __restrict__ energy) {
    __shared__ float red[64];
    int atom = blockIdx.x, t = threadIdx.x;
    float a = 0.0f;
    for (int q = t; q < 240; q += 64) a += f3[(size_t)atom * 240 + q] * fw4[q];
    red[t] = a;
    __syncthreads();
    for (int s = 32; s > 0; s >>= 1) {
        if (t < s) red[t] += red[t + s];
        __syncthreads();
    }
    if (t == 0) {
        float ae = red[0] + fb4[0] + bias_atom[atype[atom]];
        atomicAdd(&energy[atom / NLOC_], ae);
    }
}

// d_pre3 = fw4[j] * (1 - (f3-f2)^2)       (dE/dae == 1 per atom)
__global__ void kdp3(const float* __restrict__ f3, const float* __restrict__ f2,
                     const float* __restrict__ fw4, float* __restrict__ dp3) {
    int i = blockIdx.x * blockDim.x + threadIdx.x;
    if (i >= NA_ * 240) return;
    float tt = f3[i] - f2[i];
    dp3[i] = fw4[i % 240] * (1.0f - tt * tt);
}

// ----------------------------------------------------------- dD -> dGR
__global__ void kdgr(const float* __restrict__ dD, const float* __restrict__ GR,
                     float* __restrict__ dGR) {
    __shared__ float sD[1600];
    __shared__ float sG[400];
    int atom = blockIdx.x, t = threadIdx.x;
    for (int q = t; q < 1600; q += 128) sD[q] = dD[(size_t)atom * 1600 + q];
    for (int q = t; q < 400; q += 128) sG[q] = GR[(size_t)atom * 400 + q];
    __syncthreads();
    for (int idx = t; idx < 400; idx += 128) {
        int c = idx / 100, mm = idx % 100;
        float a = 0.0f;
        #pragma unroll
        for (int l = 0; l < 16; ++l) a += sD[mm * 16 + l] * sG[c * 100 + l];
        if (mm < 16)
            for (int m2 = 0; m2 < 100; ++m2) a += sD[m2 * 16 + mm] * sG[c * 100 + m2];
        dGR[(size_t)atom * 400 + idx] = a;
    }
}

// --------------------------- backward through embedding + geometry -> fe
__global__ void kbwd(const float* __restrict__ ext, const int* __restrict__ selected,
                     const float* __restrict__ dGR,
                     const float* __restrict__ ew1, const float* __restrict__ eb1,
                     const float* __restrict__ ew2, const float* __restrict__ eb2,
                     const float* __restrict__ ew3, const float* __restrict__ eb3,
                     float* __restrict__ fe) {
    __shared__ float sdg[400];
    __shared__ float cred[3][64];
    __shared__ float sw3[50 * 100];
    __shared__ float sw2[25 * 50];
    __shared__ float sw1[25], sb1[25], sb2[50], sb3[100];

    int atom = blockIdx.x;
    int b = atom / NLOC_, i = atom % NLOC_;
    int t = threadIdx.x;
    for (int q = t; q < 400; q += 64) sdg[q] = dGR[(size_t)atom * 400 + q];
    for (int q = t; q < 5000; q += NNEI_) sw3[q] = ew3[q];
    for (int q = t; q < 1250; q += NNEI_) sw2[q] = ew2[q];
    if (t < 25) { sw1[t] = ew1[t]; sb1[t] = eb1[t]; }
    if (t < 50) sb2[t] = eb2[t];
    for (int q = t; q < 100; q += NNEI_) sb3[q] = eb3[q];
    __syncthreads();

    int j = selected[(size_t)atom * NNEI_ + t];
    float cx = ext[((size_t)b * NALL_ + i) * 3 + 0];
    float cy = ext[((size_t)b * NALL_ + i) * 3 + 1];
    float cz = ext[((size_t)b * NALL_ + i) * 3 + 2];
    float x = ext[((size_t)b * NALL_ + j) * 3 + 0] - cx;
    float y = ext[((size_t)b * NALL_ + j) * 3 + 1] - cy;
    float z = ext[((size_t)b * NALL_ + j) * 3 + 2] - cz;
    float r = sqrtf(x * x + y * y + z * z + EPS_);
    float dsw;
    float s = sw_fn(r, &dsw);
    float invr = 1.0f / r;
    float Rm1 = s * x * invr, Rm2 = s * y * invr, Rm3 = s * z * invr;

    // recompute embedding forward (cheaper than storing 185 MB of activations)
    float h1[25], h2[50];
    #pragma unroll
    for (int a = 0; a < 25; ++a) h1[a] = tanhf(s * sw1[a] + sb1[a]);
    for (int a = 0; a < 50; ++a) {
        float p = sb2[a];
        #pragma unroll
        for (int q = 0; q < 25; ++q) p += h1[q] * sw2[q * 50 + a];
        h2[a] = tanhf(p) + h1[a % 25];
    }

    const float invn = 1.0f / NNEI_;
    float dh2[50];
    #pragma unroll
    for (int a = 0; a < 50; ++a) dh2[a] = 0.0f;
    float dRm0 = 0.f, dRm1 = 0.f, dRm2 = 0.f, dRm3 = 0.f;

    // stream layer-3 backward (g is never materialized)
    for (int jg = 0; jg < 100; ++jg) {
        float p = sb3[jg];
        #pragma unroll
        for (int q = 0; q < 50; ++q) p += h2[q] * sw3[q * 100 + jg];
        float tg = tanhf(p);
        float gj = tg + h2[jg % 50];
        float d0 = sdg[jg], d1 = sdg[100 + jg], d2 = sdg[200 + jg], d3 = sdg[300 + jg];
        float dgj = (s * d0 + Rm1 * d1 + Rm2 * d2 + Rm3 * d3) * invn; // Rm0 == s
        dRm0 += d0 * gj * invn;
        dRm1 += d1 * gj * invn;
        dRm2 += d2 * gj * invn;
        dRm3 += d3 * gj * invn;
        float dpre = dgj * (1.0f - tg * tg);
        #pragma unroll
        for (int q = 0; q < 50; ++q) dh2[q] += dpre * sw3[q * 100 + jg];
        dh2[jg % 50] += dgj;
    }

    float dh1[25];
    #pragma unroll
    for (int a = 0; a < 25; ++a) dh1[a] = 0.0f;
    for (int i2 = 0; i2 < 50; ++i2) {
        float th = h2[i2] - h1[i2 % 25];
        float dpre = dh2[i2] * (1.0f - th * th);
        #pragma unroll
        for (int q = 0; q < 25; ++q) dh1[q] += dpre * sw2[q * 50 + i2];
        dh1[i2 % 25] += dh2[i2];
    }
    float dsemb = 0.0f;
    #pragma unroll
    for (int q = 0; q < 25; ++q) dsemb += dh1[q] * (1.0f - h1[q] * h1[q]) * sw1[q];

    // geometry backward:  dE/du_d = dEdr * u_d/r + s * dRm_d / r
    float dotru = dRm1 * x + dRm2 * y + dRm3 * z;
    float Ws   = dRm0 + dsemb + dotru * invr;
    float dEdr = Ws * dsw - dotru * s * invr * invr;
    float cmn  = dEdr * invr;
    float du0 = cmn * x + s * invr * dRm1;
    float du1 = cmn * y + s * invr * dRm2;
    float du2 = cmn * z + s * invr * dRm3;

    atomicAdd(&fe[((size_t)b * NALL_ + j) * 3 + 0], du0);
    atomicAdd(&fe[((size_t)b * NALL_ + j) * 3 + 1], du1);
    atomicAdd(&fe[((size_t)b * NALL_ + j) * 3 + 2], du2);

    // reduce the -du center contribution across the 64 neighbors first
    cred[0][t] = -du0; cred[1][t] = -du1; cred[2][t] = -du2;
    __syncthreads();
    for (int sft = 32; sft > 0; sft >>= 1) {
        if (t < sft) {
            cred[0][t] += cred[0][t + sft];
            cred[1][t] += cred[1][t + sft];
            cred[2][t] += cred[2][t + sft];
        }
        __syncthreads();
    }
    if (t == 0) {
        atomicAdd(&fe[((size_t)b * NALL_ + i) * 3 + 0], cred[0][0]);
        atomicAdd(&fe[((size_t)b * NALL_ + i) * 3 + 1], cred[1][0]);
        atomicAdd(&fe[((size_t)b * NALL_ + i) * 3 + 2], cred[2][0]);
    }
}

// ------------------------------------------------------- virial and force
__global__ void kvirial(const float* __restrict__ ext, const float* __restrict__ fe,
                        float* __restrict__ vir) {
    __shared__ float acc[2][9];
    int t = threadIdx.x;
    if (t < 18) acc[t / 9][t % 9] = 0.0f;
    __syncthreads();
    int idx = blockIdx.x * blockDim.x + t;
    if (idx < B_ * NALL_) {
        int b = idx / NALL_;
        float e0 = ext[idx * 3 + 0], e1 = ext[idx * 3 + 1], e2 = ext[idx * 3 + 2];
        float f0 = fe[idx * 3 + 0], f1 = fe[idx * 3 + 1], f2 = fe[idx * 3 + 2];
        atomicAdd(&acc[b][0], -e0 * f0); atomicAdd(&acc[b][1], -e0 * f1); atomicAdd(&acc[b][2], -e0 * f2);
        atomicAdd(&acc[b][3], -e1 * f0); atomicAdd(&acc[b][4], -e1 * f1); atomicAdd(&acc[b][5], -e1 * f2);
        atomicAdd(&acc[b][6], -e2 * f0); atomicAdd(&acc[b][7], -e2 * f1); atomicAdd(&acc[b][8], -e2 * f2);
    }
    __syncthreads();
    if (t < 18) atomicAdd(&vir[t], acc[t / 9][t % 9]);
}

__global__ void kforce(const float* __restrict__ fe, const int* __restrict__ mapping,
                       float* __restrict__ force) {
    int idx = blockIdx.x * blockDim.x + threadIdx.x;
    if (idx >= B_ * NALL_) return;
    int b = idx / NALL_;
    int m = mapping[idx];
    #pragma unroll
    for (int c = 0; c < 3; ++c)
        atomicAdd(&force[((size_t)b * NLOC_ + m) * 3 + c], -fe[idx * 3 + c]);
}

// ----------------------------------------------------------------- launch
extern "C" void kernel_launch(void* const* d_in, const int* in_sizes, int n_in,
                              void* d_out, int out_size, void* d_ws, size_t ws_size,
                              hipStream_t stream) {
    const float* coord = (const float*)d_in[0];
    const float* shift = (const float*)d_in[1];
    const float* ew1 = (const float*)d_in[3];
    const float* eb1 = (const float*)d_in[4];
    const float* ew2 = (const float*)d_in[5];
    const float* eb2 = (const float*)d_in[6];
    const float* ew3 = (const float*)d_in[7];
    const float* eb3 = (const float*)d_in[8];
    const float* fw1 = (const float*)d_in[9];
    const float* fb1 = (const float*)d_in[10];
    const float* fw2 = (const float*)d_in[11];
    const float* fb2 = (const float*)d_in[12];
    const float* fw3 = (const float*)d_in[13];
    const float* fb3 = (const float*)d_in[14];
    const float* fw4 = (const float*)d_in[15];
    const float* fb4 = (const float*)d_in[16];
    const float* bias_atom = (const float*)d_in[17];
    const int* atype = (const int*)d_in[18];
    const int* mapping = (const int*)d_in[20];
    const int* selected = (const int*)d_in[21];

    float* ws = (float*)d_ws;
    size_t off = 0;
    float* ext  = ws + off; off += (size_t)B_ * NALL_ * 3;
    float* Dbuf = ws + off; off += (size_t)NA_ * 1600;   // D forward, reused as dD backward
    float* GR   = ws + off; off += (size_t)NA_ * 400;
    float* f1   = ws + off; off += (size_t)NA_ * 240;
    float* f2   = ws + off; off += (size_t)NA_ * 240;
    float* f3   = ws + off; off += (size_t)NA_ * 240;
    float* dp3  = ws + off; off += (size_t)NA_ * 240;
    float* dp2  = ws + off; off += (size_t)NA_ * 240;
    float* dp1  = ws + off; off += (size_t)NA_ * 240;
    float* dGRb = ws + off; off += (size_t)NA_ * 400;
    float* fe   = ws + off; off += (size_t)B_ * NALL_ * 3;

    float* energy = (float*)d_out;                 // [B]
    float* force  = energy + B_;                   // [B,NLOC,3]
    float* vir    = force + (size_t)B_ * NLOC_ * 3;// [B,3,3]
    const int nout = B_ + B_ * NLOC_ * 3 + B_ * 9;

    kzero<<<(nout + 255) / 256, 256, 0, stream>>>(energy, nout);
    kzero<<<(B_ * NALL_ * 3 + 255) / 256, 256, 0, stream>>>(fe, B_ * NALL_ * 3);

    kext<<<(B_ * NALL_ + 255) / 256, 256, 0, stream>>>(coord, shift, mapping, ext);
    kenv<<<NA_, NNEI_, 0, stream>>>(ext, selected, ew1, eb1, ew2, eb2, ew3, eb3, GR, Dbuf);

    dim3 gemmBlk(32, 4);                           // 4 waves / block, 1 tile / wave
    int tilesN240  = (NA_ / 16) * (240 / 16);
    int tilesN1600 = (NA_ / 16) * (1600 / 16);
    int g240  = (tilesN240 + 3) / 4;
    int g1600 = (tilesN1600 + 3) / 4;

    // fitting net forward (fp32 WMMA, fused bias/tanh/resnet epilogues)
    wmma_gemm<false, 1><<<g240, gemmBlk, 0, stream>>>(Dbuf, fw1, f1, fb1, nullptr, nullptr, nullptr, nullptr,
                                                      NA_, 240, 1600);
    wmma_gemm<false, 2><<<g240, gemmBlk, 0, stream>>>(f1, fw2, f2, fb2, f1, nullptr, nullptr, nullptr,
                                                      NA_, 240, 240);
    wmma_gemm<false, 2><<<g240, gemmBlk, 0, stream>>>(f2, fw3, f3, fb3, f2, nullptr, nullptr, nullptr,
                                                      NA_, 240, 240);

    kenergy<<<NA_, 64, 0, stream>>>(f3, fw4, fb4, bias_atom, atype, energy);

    // fitting net backward (transposed-B WMMA, fused tanh' epilogues)
    kdp3<<<(NA_ * 240 + 255) / 256, 256, 0, stream>>>(f3, f2, fw4, dp3);
    wmma_gemm<true, 3><<<g240, gemmBlk, 0, stream>>>(dp3, fw3, dp2, nullptr, nullptr, f2, f1, fw4,
                                                     NA_, 240, 240);
    wmma_gemm<true, 3><<<g240, gemmBlk, 0, stream>>>(dp2, fw2, dp1, nullptr, nullptr, f1, nullptr, nullptr,
                                                     NA_, 240, 240);
    wmma_gemm<true, 0><<<g1600, gemmBlk, 0, stream>>>(dp1, fw1, Dbuf, nullptr, nullptr, nullptr, nullptr, nullptr,
                                                      NA_, 1600, 240);   // dD

    kdgr<<<NA_, 128, 0, stream>>>(Dbuf, GR, dGRb);
    kbwd<<<NA_, NNEI_, 0, stream>>>(ext, selected, dGRb, ew1, eb1, ew2, eb2, ew3, eb3, fe);

    kvirial<<<(B_ * NALL_ + 255) / 256, 256, 0, stream>>>(ext, fe, vir);
    kforce<<<(B_ * NALL_ + 255) / 256, 256, 0, stream>>>(fe, mapping, force);
}